// QuantumTransformerVQC_12077448036881
// MI455X (gfx1250) — compile-verified
//
#include <hip/hip_runtime.h>
#include <math.h>

// ---------------------------------------------------------------------------
// QuantumTransformerVQC on MI455X (gfx1250), wave32, WMMA f32 16x16x4.
//
// Each VQC circuit (after state-prep) is a fixed unitary U depending only on
// weights.  We precompute the real embedding A = [Re U; Im U] once per call
// (setup kernel), then every VQC evaluation is a 16xK f32 GEMM over a tile of
// 16 samples using V_WMMA_F32_16X16X4_F32.
// ---------------------------------------------------------------------------

typedef float v2f __attribute__((ext_vector_type(2)));
typedef float v8f __attribute__((ext_vector_type(8)));

#define PST 33  // LDS row stride for psi staging (gcd(33,64)=1 -> no bank conflicts)

static __device__ __forceinline__ v8f wmma4(v2f a, v2f b, v8f c) {
  // 8-arg pattern: (neg_a, A, neg_b, B, c_mod, C, reuse_a, reuse_b)
  return __builtin_amdgcn_wmma_f32_16x16x4_f32(false, a, false, b, (short)0, c,
                                               false, false);
}

// ---------------------------------------------------------------------------
// Setup: build total-circuit unitaries column by column.
// vr/vi hold one column (length 2^n complex) of U; gates applied in order.
// ---------------------------------------------------------------------------
static __device__ void run_circuit(float* vr, float* vi, int n, int depth,
                                   const float* w, int lstride) {
  const int N = 1 << n;
  for (int l = 0; l < depth; ++l) {
    for (int q = 0; q < n; ++q) {
      const float* wr = w + l * lstride + q * 3;
      float phi = wr[0], th = wr[1], om = wr[2];
      float ct = cosf(0.5f * th), st = sinf(0.5f * th);
      float a = 0.5f * (phi + om), bb = 0.5f * (phi - om);
      float epr = cosf(a), epi = -sinf(a);   // exp(-0.5i(phi+om))
      float emr = cosf(bb), emi = sinf(bb);  // exp(+0.5i(phi-om))
      // row0 = [ep*c, -em*s]; row1 = [conj(em)*s, conj(ep)*c]
      float g00r = epr * ct, g00i = epi * ct;
      float g01r = -emr * st, g01i = -emi * st;
      float g10r = emr * st, g10i = -emi * st;
      float g11r = epr * ct, g11i = -epi * ct;
      int mask = 1 << (n - 1 - q);
      for (int i = 0; i < N; ++i) {
        if (i & mask) continue;
        int j = i | mask;
        float xr = vr[i], xi = vi[i], yr = vr[j], yi = vi[j];
        vr[i] = g00r * xr - g00i * xi + g01r * yr - g01i * yi;
        vi[i] = g00r * xi + g00i * xr + g01r * yi + g01i * yr;
        vr[j] = g10r * xr - g10i * xi + g11r * yr - g11i * yi;
        vi[j] = g10r * xi + g10i * xr + g11r * yi + g11i * yr;
      }
    }
    int r = l % (n - 1) + 1;
    for (int q = 0; q < n; ++q) {
      int ctrl = q, tgt = (q + r) % n;
      int cmask = 1 << (n - 1 - ctrl), tmask = 1 << (n - 1 - tgt);
      float tr[32], ti[32];
      for (int i = 0; i < N; ++i) {
        int src = (i & cmask) ? (i ^ tmask) : i;
        tr[i] = vr[src];
        ti[i] = vi[src];
      }
      for (int i = 0; i < N; ++i) { vr[i] = tr[i]; vi[i] = ti[i]; }
    }
  }
}

// 45 3-qubit instances -> A3[inst][16][8]; 1 5-qubit instance -> A5[64][32]
extern "C" __global__ void qvt_setup(const float* __restrict__ thQ,
                                     const float* __restrict__ thK,
                                     const float* __restrict__ thV,
                                     const float* __restrict__ thF,
                                     const float* __restrict__ thR,
                                     const float* __restrict__ thFin,
                                     float* __restrict__ A3,
                                     float* __restrict__ A5) {
  int tid = blockIdx.x * blockDim.x + threadIdx.x;
  if (tid < 360) {  // 45 instances x 8 columns
    int inst = tid >> 3;
    int col = tid & 7;
    const float* wbase;
    if (inst < 40) {
      int l = inst / 20, rem = inst % 20, kind = rem / 5, b = rem % 5;
      const float* th = (kind == 0) ? thQ : (kind == 1) ? thK
                        : (kind == 2) ? thV : thF;
      wbase = th + (l * 5 + b) * 18;  // (L,5,D,3,3) layout
    } else {
      wbase = thR + (inst - 40) * 18;  // (5,D,3,3)
    }
    float vr[8], vi[8];
    for (int i = 0; i < 8; ++i) { vr[i] = (i == col) ? 1.f : 0.f; vi[i] = 0.f; }
    run_circuit(vr, vi, 3, 2, wbase, 9);
    float* A = A3 + inst * 128;
    for (int j = 0; j < 8; ++j) {
      A[j * 8 + col] = vr[j];        // Re rows 0..7
      A[(j + 8) * 8 + col] = vi[j];  // Im rows 8..15
    }
  } else if (tid < 392) {  // 32 columns of 5-qubit circuit
    int col = tid - 360;
    float vr[32], vi[32];
    for (int i = 0; i < 32; ++i) { vr[i] = (i == col) ? 1.f : 0.f; vi[i] = 0.f; }
    run_circuit(vr, vi, 5, 2, thFin, 15);  // (D,5,3)
    for (int j = 0; j < 32; ++j) {
      A5[j * 32 + col] = vr[j];         // Re rows 0..31
      A5[(j + 32) * 32 + col] = vi[j];  // Im rows 32..63
    }
  }
}

// ---------------------------------------------------------------------------
// Cooperative 3-qubit VQC over the wave's 32 samples (two 16-sample N-tiles).
// Aw: 16x8 row-major embedded unitary.  Writes 3 outputs per sample into
// outl[sample][col..col+2] (LDS, row stride 15).
// ---------------------------------------------------------------------------
static __device__ __forceinline__ void vqc3(const float* __restrict__ Aw,
                                            float h0, float h1, float h2,
                                            float* psi, float* outl, int col,
                                            int L) {
  float c0 = cosf(0.5f * h0), s0 = sinf(0.5f * h0);
  float c1 = cosf(0.5f * h1), s1 = sinf(0.5f * h1);
  float c2 = cosf(0.5f * h2), s2 = sinf(0.5f * h2);
  float* pr = psi + L * PST;  // qubit0 is MSB of the 3-bit amp index
  pr[0] = c0 * c1 * c2; pr[1] = c0 * c1 * s2;
  pr[2] = c0 * s1 * c2; pr[3] = c0 * s1 * s2;
  pr[4] = s0 * c1 * c2; pr[5] = s0 * c1 * s2;
  pr[6] = s0 * s1 * c2; pr[7] = s0 * s1 * s2;
  __syncthreads();
  const int half = L >> 4, m = L & 15, ko = 2 * half;
  // A frag (16x4): lane m holds A[m][K], A[m][K+1]; high half-wave K+=2
  v2f A0 = {Aw[m * 8 + ko], Aw[m * 8 + ko + 1]};
  v2f A1 = {Aw[m * 8 + 4 + ko], Aw[m * 8 + 4 + ko + 1]};
#pragma unroll
  for (int t = 0; t < 2; ++t) {
    const float* prt = psi + (t * 16 + m) * PST;
    v2f B0 = {prt[ko], prt[ko + 1]};
    v2f B1 = {prt[4 + ko], prt[4 + ko + 1]};
    v8f c = {0.f, 0.f, 0.f, 0.f, 0.f, 0.f, 0.f, 0.f};
    c = wmma4(A0, B0, c);
    c = wmma4(A1, B1, c);
    // lanes 0-15: c[r] = Re(amp r) of sample (L&15); lanes 16-31: Im(amp r)
    float o0 = 0.f, o1 = 0.f, o2 = 0.f;
#pragma unroll
    for (int r = 0; r < 8; ++r) {
      float p = c[r] * c[r];
      o0 += ((r >> 2) & 1) ? -p : p;
      o1 += ((r >> 1) & 1) ? -p : p;
      o2 += (r & 1) ? -p : p;
    }
    o0 += __shfl_xor(o0, 16, 32);  // Re^2 + Im^2 across half-waves
    o1 += __shfl_xor(o1, 16, 32);
    o2 += __shfl_xor(o2, 16, 32);
    if (L < 16) {
      float* orow = outl + (t * 16 + L) * 15 + col;
      orow[0] = o0; orow[1] = o1; orow[2] = o2;
    }
  }
  __syncthreads();
}

static __device__ __forceinline__ void layer_norm15(float* H, const float* add,
                                                    const float* g,
                                                    const float* b) {
  float y[15], mean = 0.f;
#pragma unroll
  for (int i = 0; i < 15; ++i) { y[i] = H[i] + add[i]; mean += y[i]; }
  mean *= (1.f / 15.f);
  float var = 0.f;
#pragma unroll
  for (int i = 0; i < 15; ++i) { float d = y[i] - mean; var += d * d; }
  var *= (1.f / 15.f);
  float r = rsqrtf(var + 1e-5f);
#pragma unroll
  for (int i = 0; i < 15; ++i) H[i] = (y[i] - mean) * r * g[i] + b[i];
}

extern "C" __global__ void __launch_bounds__(32)
qvt_main(const float* __restrict__ x, const float* __restrict__ attn_gamma,
         const float* __restrict__ attn_beta, const float* __restrict__ ffn_gamma,
         const float* __restrict__ ffn_beta, const float* __restrict__ A3,
         const float* __restrict__ A5, float* __restrict__ out) {
  __shared__ float psi[32 * PST];
  __shared__ float stq[32 * 15];
  __shared__ float stk[32 * 15];
  __shared__ float stv[32 * 15];
  const int L = threadIdx.x;
  const int sbase = blockIdx.x * 32;

  float H[15];
  H[0] = 0.f; H[14] = 0.f;
#pragma unroll
  for (int i = 0; i < 13; ++i) H[i + 1] = x[(sbase + L) * 13 + i];

#pragma unroll 1
  for (int l = 0; l < 2; ++l) {
    float Hc[15];
#pragma unroll
    for (int i = 0; i < 15; ++i) Hc[i] = H[i];
#pragma unroll
    for (int b = 0; b < 5; ++b)
      vqc3(A3 + (l * 20 + b) * 128, Hc[3 * b], Hc[3 * b + 1], Hc[3 * b + 2],
           psi, stq, 3 * b, L);
#pragma unroll
    for (int b = 0; b < 5; ++b)
      vqc3(A3 + (l * 20 + 5 + b) * 128, Hc[3 * b], Hc[3 * b + 1],
           Hc[3 * b + 2], psi, stk, 3 * b, L);
#pragma unroll
    for (int b = 0; b < 5; ++b)
      vqc3(A3 + (l * 20 + 10 + b) * 128, Hc[3 * b], Hc[3 * b + 1],
           Hc[3 * b + 2], psi, stv, 3 * b, L);

    // ---- attention (per-lane scalar; lane L owns sample L) ----
    float ao[15];
    const float rs3 = 0.57735026918962576f;  // 1/sqrt(3)
#pragma unroll
    for (int t = 0; t < 5; ++t) {
      float sc[5], mx = -1e30f;
#pragma unroll
      for (int u = 0; u < 5; ++u) {
        float d = stq[L * 15 + 3 * t] * stk[L * 15 + 3 * u] +
                  stq[L * 15 + 3 * t + 1] * stk[L * 15 + 3 * u + 1] +
                  stq[L * 15 + 3 * t + 2] * stk[L * 15 + 3 * u + 2];
        sc[u] = d * rs3;
        mx = fmaxf(mx, sc[u]);
      }
      float se = 0.f;
#pragma unroll
      for (int u = 0; u < 5; ++u) { sc[u] = expf(sc[u] - mx); se += sc[u]; }
      float inv = 1.f / se;
#pragma unroll
      for (int d = 0; d < 3; ++d) {
        float acc = 0.f;
#pragma unroll
        for (int u = 0; u < 5; ++u) acc += sc[u] * stv[L * 15 + 3 * u + d];
        ao[3 * t + d] = acc * inv;
      }
    }
    layer_norm15(H, ao, attn_gamma + l * 15, attn_beta + l * 15);
    __syncthreads();

    // ---- FFN ----
#pragma unroll
    for (int b = 0; b < 5; ++b)
      vqc3(A3 + (l * 20 + 15 + b) * 128, H[3 * b], H[3 * b + 1], H[3 * b + 2],
           psi, stq, 3 * b, L);
    float fo[15];
#pragma unroll
    for (int i = 0; i < 15; ++i) fo[i] = stq[L * 15 + i];
    layer_norm15(H, fo, ffn_gamma + l * 15, ffn_beta + l * 15);
    __syncthreads();
  }

  // ---- reduce: 5 VQCs, keep q=0 output only ----
#pragma unroll
  for (int b = 0; b < 5; ++b)
    vqc3(A3 + (40 + b) * 128, H[3 * b], H[3 * b + 1], H[3 * b + 2], psi, stq,
         3 * b, L);
  float h5[5];
#pragma unroll
  for (int b = 0; b < 5; ++b) h5[b] = stq[L * 15 + 3 * b];
  __syncthreads();

  // ---- final 5-qubit VQC: D(64x16) = A5(64x32) @ Psi(32x16) per tile ----
  float cc[5], sn[5];
#pragma unroll
  for (int q = 0; q < 5; ++q) {
    cc[q] = cosf(0.5f * h5[q]);
    sn[q] = sinf(0.5f * h5[q]);
  }
  {
    float* pr = psi + L * PST;
#pragma unroll
    for (int i = 0; i < 32; ++i) {
      float p = ((i >> 4) & 1) ? sn[0] : cc[0];
      p *= ((i >> 3) & 1) ? sn[1] : cc[1];
      p *= ((i >> 2) & 1) ? sn[2] : cc[2];
      p *= ((i >> 1) & 1) ? sn[3] : cc[3];
      p *= (i & 1) ? sn[4] : cc[4];
      pr[i] = p;
    }
  }
  __syncthreads();

  const int half = L >> 4, m = L & 15, ko = 2 * half;
  float res[2];
#pragma unroll 1
  for (int t = 0; t < 2; ++t) {
    v8f d0 = {0.f, 0.f, 0.f, 0.f, 0.f, 0.f, 0.f, 0.f};
    v8f d1 = d0, d2 = d0, d3 = d0;
    const float* prt = psi + (t * 16 + m) * PST;
#pragma unroll
    for (int ks = 0; ks < 8; ++ks) {
      v2f B = {prt[ks * 4 + ko], prt[ks * 4 + ko + 1]};
      v2f a0 = {A5[m * 32 + ks * 4 + ko], A5[m * 32 + ks * 4 + ko + 1]};
      v2f a1 = {A5[(16 + m) * 32 + ks * 4 + ko],
                A5[(16 + m) * 32 + ks * 4 + ko + 1]};
      v2f a2 = {A5[(32 + m) * 32 + ks * 4 + ko],
                A5[(32 + m) * 32 + ks * 4 + ko + 1]};
      v2f a3 = {A5[(48 + m) * 32 + ks * 4 + ko],
                A5[(48 + m) * 32 + ks * 4 + ko + 1]};
      d0 = wmma4(a0, B, d0);  // Re, amps 0..15  -> sign +1
      d1 = wmma4(a1, B, d1);  // Re, amps 16..31 -> sign -1
      d2 = wmma4(a2, B, d2);  // Im, amps 0..15  -> sign +1
      d3 = wmma4(a3, B, d3);  // Im, amps 16..31 -> sign -1
    }
    float o = 0.f;
#pragma unroll
    for (int r = 0; r < 8; ++r)
      o += d0[r] * d0[r] + d2[r] * d2[r] - d1[r] * d1[r] - d3[r] * d3[r];
    o += __shfl_xor(o, 16, 32);
    res[t] = o;
  }
  const float MULTF = (float)(3.14159265358979323846 - 1.1920928955078125e-07);
  if (L < 16) {
    out[sbase + L] = res[0] * MULTF;
    out[sbase + 16 + L] = res[1] * MULTF;
  }
}

// ---------------------------------------------------------------------------
extern "C" void kernel_launch(void* const* d_in, const int* in_sizes, int n_in,
                              void* d_out, int out_size, void* d_ws,
                              size_t ws_size, hipStream_t stream) {
  const float* x = (const float*)d_in[0];
  const float* thQ = (const float*)d_in[1];
  const float* thK = (const float*)d_in[2];
  const float* thV = (const float*)d_in[3];
  const float* thF = (const float*)d_in[4];
  const float* ag = (const float*)d_in[5];
  const float* ab = (const float*)d_in[6];
  const float* fg = (const float*)d_in[7];
  const float* fb = (const float*)d_in[8];
  const float* thR = (const float*)d_in[9];
  const float* thFin = (const float*)d_in[10];

  float* A3 = (float*)d_ws;           // 45 * 16 * 8 floats
  float* A5 = A3 + 45 * 128;          // 64 * 32 floats

  qvt_setup<<<4, 128, 0, stream>>>(thQ, thK, thV, thF, thR, thFin, A3, A5);

  int nsamp = in_sizes[0] / 13;       // 65536
  qvt_main<<<nsamp / 32, 32, 0, stream>>>(x, ag, ab, fg, fb, A3, A5,
                                          (float*)d_out);
}